// ConstraintAndResult_45363444580568
// MI455X (gfx1250) — compile-verified
//
#include <hip/hip_runtime.h>

namespace {
constexpr int Lc     = 1024;
constexpr int Bc     = 32;
constexpr int NTc    = 16;   // 1024 / 64 tiles per dimension
constexpr int STEPSc = 20;
}

typedef float v2f __attribute__((ext_vector_type(2)));
typedef float v8f __attribute__((ext_vector_type(8)));

__device__ __forceinline__ float sgnf(float x) {
    return (x > 0.f) ? 1.f : ((x < 0.f) ? -1.f : 0.f);
}

// Row-sum of a 64x64 tile held in LDS (row-major, pitch 65) for row group
// [R0, R0+16) using V_WMMA_F32_16X16X4_F32 with an all-ones B operand.
// A-operand layout (32-bit A 16x4): lane l<16 holds row l, K={0,1};
// lane l+16 holds row l, K={2,3}. D layout: row m -> VGPR m (lanes 0-15)
// and row m+8 -> VGPR m (lanes 16-31); row sum replicated across N.
__device__ __forceinline__ void tile_rowsum_wmma(const float* __restrict__ buf,
                                                 int R0,
                                                 float* __restrict__ dst)
{
    const int lane = threadIdx.x & 31;
    const int l16  = lane & 15;
    const int hi   = lane >> 4;
    v8f acc = {};
    v2f ones; ones[0] = 1.f; ones[1] = 1.f;
#pragma unroll
    for (int q = 0; q < 16; ++q) {
        const int k0 = 4 * q + 2 * hi;
        v2f a;
        a[0] = buf[(R0 + l16) * 65 + k0];
        a[1] = buf[(R0 + l16) * 65 + k0 + 1];
        acc = __builtin_amdgcn_wmma_f32_16x16x4_f32(false, a, false, ones,
                                                    (short)0, acc, false, false);
    }
    if (lane == 0) {
#pragma unroll
        for (int m = 0; m < 8; ++m) dst[m] = acc[m];
    } else if (lane == 16) {
#pragma unroll
        for (int m = 0; m < 8; ++m) dst[8 + m] = acc[m];
    }
}

// ---------------------------------------------------------------------------
// Init: Ahat = scores (copy), partial[b][cb][i] = 64-col chunk row sums of
// A0 = 0.5*(sc + sc^T)*M.
// ---------------------------------------------------------------------------
__global__ __launch_bounds__(256)
void rna_init_kernel(const float* __restrict__ scores,
                     const float* __restrict__ Mg,
                     float* __restrict__ Ahat,
                     float* __restrict__ partial)
{
    const int bi = blockIdx.x, bj = blockIdx.y, b = blockIdx.z;
    if (bj < bi) return;
    const bool diag = (bi == bj);
    const int i0 = bi * 64, j0 = bj * 64;
    const int tid = threadIdx.x;
    const int c0 = tid & 63;
    const int rb = tid >> 6;
    const size_t bo = (size_t)b * Lc * Lc;

    __shared__ float BufA[64 * 65];   // BufA[X*65+Y] = scores[j0+Y][i0+X]
    __shared__ float BufS[64 * 65];   // BufS[X*65+Y] = scores[i0+X][j0+Y]

    float scI[16], mij[16], mji[16];

#pragma unroll
    for (int k = 0; k < 16; ++k) {                 // tile (j,i): load + copy
        const int r = rb + 4 * k;
        const size_t gi = bo + (size_t)(j0 + r) * Lc + (i0 + c0);
        const float v = scores[gi];
        BufA[c0 * 65 + r] = v;
        Ahat[gi] = v;
    }
#pragma unroll
    for (int k = 0; k < 16; ++k) {                 // tile (i,j): load + copy
        const int r = rb + 4 * k;
        const size_t gi = bo + (size_t)(i0 + r) * Lc + (j0 + c0);
        const float v = scores[gi];
        scI[k] = v;
        BufS[r * 65 + c0] = v;
        if (!diag) Ahat[gi] = v;
        mij[k] = Mg[gi];
        if (!diag) mji[k] = Mg[bo + (size_t)(j0 + r) * Lc + (i0 + c0)];
    }
    __syncthreads();

    float aij[16], aji[16];
#pragma unroll
    for (int k = 0; k < 16; ++k) {
        const int r = rb + 4 * k;
        const float y1 = BufA[r * 65 + c0];        // sc[j0+c0][i0+r]
        aij[k] = 0.5f * (scI[k] + y1) * mij[k];
        if (!diag) {
            const float y2 = BufA[c0 * 65 + r];    // sc[j0+r][i0+c0]
            const float y3 = BufS[c0 * 65 + r];    // sc[i0+c0][j0+r]
            aji[k] = 0.5f * (y2 + y3) * mji[k];
        }
    }
    __syncthreads();
#pragma unroll
    for (int k = 0; k < 16; ++k) {                 // marshal A tiles for WMMA
        const int r = rb + 4 * k;
        BufA[r * 65 + c0] = aij[k];
        if (!diag) BufS[r * 65 + c0] = aji[k];
    }
    __syncthreads();

    const int w = tid >> 5;
    const bool isIJ = (w < 4);
    if (!diag || isIJ) {
        const float* buf = isIJ ? BufA : BufS;
        const int R0 = (w & 3) * 16;
        const int cb = isIJ ? bj : bi;
        const int rowbase = (isIJ ? i0 : j0) + R0;
        float* dst = partial + ((size_t)b * NTc + cb) * Lc + rowbase;
        tile_rowsum_wmma(buf, R0, dst);
    }
}

// ---------------------------------------------------------------------------
// One PGD step on the (bi,bj)/(bj,bi) tile pair + WMMA row sums of new A.
// ---------------------------------------------------------------------------
__global__ __launch_bounds__(256)
void rna_step_kernel(const float* __restrict__ scores,
                     const float* __restrict__ Mg,
                     const float* __restrict__ rho,
                     const float* __restrict__ s_p,
                     const float* __restrict__ alpha_p,
                     const float* __restrict__ lra_p,
                     float* __restrict__ Ahat,
                     const float* __restrict__ cArr,
                     float* __restrict__ partial,
                     int t, int isLast)
{
    const int bi = blockIdx.x, bj = blockIdx.y, b = blockIdx.z;
    if (bj < bi) return;
    const bool diag = (bi == bj);
    const int i0 = bi * 64, j0 = bj * 64;
    const int tid = threadIdx.x;
    const int c0 = tid & 63;
    const int rb = tid >> 6;
    const size_t bo = (size_t)b * Lc * Lc;

    const float s  = s_p[0];
    const float at = alpha_p[0] * powf(lra_p[0], (float)t);

    __shared__ float BufA[64 * 65];
    __shared__ float BufS[64 * 65];
    __shared__ float cI[64], cJ[64];

    if (tid < 64)           cI[tid]      = cArr[(size_t)b * Lc + i0 + tid];
    else if (tid < 128)     cJ[tid - 64] = cArr[(size_t)b * Lc + j0 + (tid - 64)];

#pragma unroll
    for (int k = 0; k < 16; ++k) {                 // scores (j,i) tile, transposed
        const int r = rb + 4 * k;
        BufA[c0 * 65 + r] = scores[bo + (size_t)(j0 + r) * Lc + (i0 + c0)];
    }
    __syncthreads();

    // --- update tile (i,j); BufS <- Ssym, BufA <- new Ah_ij (in place) ---
    float mij[16], ahI[16];
#pragma unroll
    for (int k = 0; k < 16; ++k) {
        const int r = rb + 4 * k;
        const size_t gi = bo + (size_t)(i0 + r) * Lc + (j0 + c0);
        const float ah = Ahat[gi];
        const float sc = scores[gi];
        const float m  = Mg[gi];
        const float rp = rho[(size_t)(i0 + r) * Lc + (j0 + c0)];
        const float ssym = 0.5f * (sc + BufA[r * 65 + c0]) - s;
        BufS[r * 65 + c0] = ssym;
        const float g2 = ssym - cI[r] - cJ[c0];
        float x = fmaf(at * ah * m, g2, ah);
        x = fmaxf(fabsf(x) - rp * at, 0.f);        // soft threshold
        x = 1.f - fmaxf(1.f - x, 0.f);             // clip above at 1
        ahI[k] = x; mij[k] = m;
        BufA[r * 65 + c0] = x;
        if (!isLast) Ahat[gi] = x;
    }
    __syncthreads();

    // --- update tile (j,i) using Ssym^T from BufS; BufS <- new Ah_ji ---
    float mji[16], ahJ[16];
    if (!diag) {
        float ssJ[16];
#pragma unroll
        for (int k = 0; k < 16; ++k) ssJ[k] = BufS[c0 * 65 + (rb + 4 * k)];
        __syncthreads();
#pragma unroll
        for (int k = 0; k < 16; ++k) {
            const int r = rb + 4 * k;
            const size_t gi = bo + (size_t)(j0 + r) * Lc + (i0 + c0);
            const float ah = Ahat[gi];
            const float m  = Mg[gi];
            const float rp = rho[(size_t)(j0 + r) * Lc + (i0 + c0)];
            const float g2 = ssJ[k] - cJ[r] - cI[c0];
            float x = fmaf(at * ah * m, g2, ah);
            x = fmaxf(fabsf(x) - rp * at, 0.f);
            x = 1.f - fmaxf(1.f - x, 0.f);
            ahJ[k] = x; mji[k] = m;
            BufS[r * 65 + c0] = x;
            if (!isLast) Ahat[gi] = x;
        }
    }
    __syncthreads();

    // --- A = 0.5*(AhN + AhN^T)*M, marshal into LDS, WMMA row sums ---
    float x1[16], x2[16];
#pragma unroll
    for (int k = 0; k < 16; ++k) {
        const int r = rb + 4 * k;
        x2[k] = BufA[c0 * 65 + r];                    // AhN_ij[c0][r]
        x1[k] = diag ? x2[k] : BufS[c0 * 65 + r];     // AhN_ji[c0][r]
    }
    __syncthreads();
#pragma unroll
    for (int k = 0; k < 16; ++k) {
        const int r = rb + 4 * k;
        const float aij = 0.5f * (ahI[k] + x1[k]) * mij[k];
        BufA[r * 65 + c0] = aij;
        if (isLast) Ahat[bo + (size_t)(i0 + r) * Lc + (j0 + c0)] = aij;
        if (!diag) {
            const float aji = 0.5f * (ahJ[k] + x2[k]) * mji[k];
            BufS[r * 65 + c0] = aji;
            if (isLast) Ahat[bo + (size_t)(j0 + r) * Lc + (i0 + c0)] = aji;
        }
    }
    __syncthreads();

    const int w = tid >> 5;
    const bool isIJ = (w < 4);
    if (!diag || isIJ) {
        const float* buf = isIJ ? BufA : BufS;
        const int R0 = (w & 3) * 16;
        const int cb = isIJ ? bj : bi;
        const int rowbase = (isIJ ? i0 : j0) + R0;
        float* dst = partial + ((size_t)b * NTc + cb) * Lc + rowbase;
        tile_rowsum_wmma(buf, R0, dst);
    }
}

// ---------------------------------------------------------------------------
// Reduce per-column-block partial row sums; init / update Lm and c = Lm*sign.
// ---------------------------------------------------------------------------
__global__ __launch_bounds__(256)
void lm_init_kernel(const float* __restrict__ partial,
                    float* __restrict__ Lm,
                    float* __restrict__ cArr,
                    const float* __restrict__ w_p)
{
    const int idx = blockIdx.x * 256 + threadIdx.x;
    if (idx >= Bc * Lc) return;
    const int b = idx >> 10, i = idx & (Lc - 1);
    float r = 0.f;
#pragma unroll
    for (int cb = 0; cb < NTc; ++cb) r += partial[((size_t)b * NTc + cb) * Lc + i];
    const float row = r - 1.f;
    const float lm = w_p[0] * fmaxf(row, 0.f);
    Lm[idx] = lm;
    cArr[idx] = lm * sgnf(row);
}

__global__ __launch_bounds__(256)
void lm_update_kernel(const float* __restrict__ partial,
                      float* __restrict__ Lm,
                      float* __restrict__ cArr,
                      const float* __restrict__ belt_p,
                      const float* __restrict__ lrb_p,
                      int t)
{
    const int idx = blockIdx.x * 256 + threadIdx.x;
    if (idx >= Bc * Lc) return;
    const int b = idx >> 10, i = idx & (Lc - 1);
    float r = 0.f;
#pragma unroll
    for (int cb = 0; cb < NTc; ++cb) r += partial[((size_t)b * NTc + cb) * Lc + i];
    const float row = r - 1.f;
    const float lm = Lm[idx] + belt_p[0] * powf(lrb_p[0], (float)t) * fmaxf(row, 0.f);
    Lm[idx] = lm;
    cArr[idx] = lm * sgnf(row);
}

// ---------------------------------------------------------------------------
extern "C" void kernel_launch(void* const* d_in, const int* in_sizes, int n_in,
                              void* d_out, int out_size, void* d_ws, size_t ws_size,
                              hipStream_t stream)
{
    (void)in_sizes; (void)n_in; (void)out_size; (void)ws_size;
    const float* scores  = (const float*)d_in[0];
    const float* Mg      = (const float*)d_in[1];
    const float* rho     = (const float*)d_in[2];
    const float* s_p     = (const float*)d_in[3];
    const float* w_p     = (const float*)d_in[4];
    const float* alpha_p = (const float*)d_in[5];
    const float* belt_p  = (const float*)d_in[6];
    const float* lra_p   = (const float*)d_in[7];
    const float* lrb_p   = (const float*)d_in[8];

    float* Ahat = (float*)d_out;                       // A_hat state lives in d_out

    float* partial = (float*)d_ws;                     // B*NT*L floats (2 MB)
    float* Lm      = partial + (size_t)Bc * NTc * Lc;  // B*L
    float* cArr    = Lm + (size_t)Bc * Lc;             // B*L

    dim3 grid(NTc, NTc, Bc), blk(256);
    rna_init_kernel<<<grid, blk, 0, stream>>>(scores, Mg, Ahat, partial);
    lm_init_kernel<<<(Bc * Lc) / 256, 256, 0, stream>>>(partial, Lm, cArr, w_p);

    for (int t = 0; t < STEPSc; ++t) {
        rna_step_kernel<<<grid, blk, 0, stream>>>(scores, Mg, rho, s_p, alpha_p, lra_p,
                                                  Ahat, cArr, partial, t,
                                                  (t == STEPSc - 1) ? 1 : 0);
        if (t < STEPSc - 1)
            lm_update_kernel<<<(Bc * Lc) / 256, 256, 0, stream>>>(partial, Lm, cArr,
                                                                  belt_p, lrb_p, t);
    }
}